// CostFn_18562848653837
// MI455X (gfx1250) — compile-verified
//
#include <hip/hip_runtime.h>

typedef __attribute__((ext_vector_type(2))) float v2f;
typedef __attribute__((ext_vector_type(8))) float v8f;

// ds_swizzle group-of-32 xor-fold: offset = (xor_mask << 10) | 0x1f
#define SWZ_XOR(v, imm) __int_as_float(__builtin_amdgcn_ds_swizzle(__float_as_int(v), (imm)))

// Wave32 sum via the CDNA5 matrix pipe.
// A holds the lane value ({v, 0}), B = ones, so D[m,n] = rowsum_m of A.
// Per the 16x16 f32 C/D layout, acc[r] = rowsum_r (lanes 0-15) and
// rowsum_{r+8} (lanes 16-31); summing acc[0..7] gives the two half-wave
// totals, and one xor-16 swizzle fold completes the wave sum.
// EXEC must be all ones at the call site (call unconditionally).
__device__ __forceinline__ float wave_reduce_wmma(float v) {
    v2f a; a[0] = v;    a[1] = 0.0f;
    v2f b; b[0] = 1.0f; b[1] = 1.0f;
    v8f acc = {};
    acc = __builtin_amdgcn_wmma_f32_16x16x4_f32(
        /*neg_a=*/false, a, /*neg_b=*/false, b,
        /*c_mod=*/(short)0, acc, /*reuse_a=*/false, /*reuse_b=*/false);
    float s = ((acc[0] + acc[1]) + (acc[2] + acc[3])) +
              ((acc[4] + acc[5]) + (acc[6] + acc[7]));
    return s + SWZ_XOR(s, 0x401f);   // fold halves: every lane = wave total
}

__device__ __forceinline__ float block_reduce(float v, float* smem) {
    const int lane = threadIdx.x & 31;
    const int wid  = threadIdx.x >> 5;
    const int nw   = (blockDim.x + 31) >> 5;
    const float ws = wave_reduce_wmma(v);   // all threads: EXEC all-1s here
    if (lane == 0) smem[wid] = ws;
    __syncthreads();
    float t = 0.0f;
    if (wid == 0) {                          // uniform per wave: wave0 fully active
        t = (lane < nw) ? smem[lane] : 0.0f;
        t += SWZ_XOR(t, 0x041f);             // xor 1
        t += SWZ_XOR(t, 0x081f);             // xor 2
        t += SWZ_XOR(t, 0x101f);             // xor 4
        t += SWZ_XOR(t, 0x201f);             // xor 8
        t += SWZ_XOR(t, 0x401f);             // xor 16
    }
    return t;                                // valid in wave 0
}

// Stage 1: one problem (b, j) per thread; reads only x[b, j, 6:12] (24 B).
// Closed-form reflected mass via Sherman-Morrison on M = 2I + 0.5 c c^T:
//   denom = 0.5*||L.s||^2 - (L.s . c)^2 / (8 + 2*||c||^2),  val = 1/denom
__global__ void reflected_mass_kernel(const float* __restrict__ x,
                                      float* __restrict__ partial, int B) {
    constexpr int T = 26, H = 1024;
    const int P = B * 26;
    const int p = blockIdx.x * blockDim.x + threadIdx.x;

    float val = 0.0f;
    if (p < P) {
        const int b = p / 26;
        const int j = p - b * 26;
        // element index b*H*T + j*T + 6 is always even -> 8-byte aligned
        const float* src = x + (size_t)b * (H * T) + (size_t)j * T + 6;
        const float2* s2 = reinterpret_cast<const float2*>(src);
        const float2 q01 = s2[0];
        const float2 q23 = s2[1];
        const float2 q45 = s2[2];
        const float q[6] = {q01.x, q01.y, q23.x, q23.y, q45.x, q45.y};

        float cq = 0.0f, n1 = 0.0f, d = 0.0f, cc = 0.0f;
#pragma unroll
        for (int k = 0; k < 6; ++k) {
            cq += q[k];
            float sv, cv;
            __sincosf(cq, &sv, &cv);          // hw v_sin_f32 / v_cos_f32
            const float Lk = 0.4f + 0.1f * (float)k;  // L = [0.4 .. 0.9]
            const float Ls = Lk * sv;
            n1 += Ls * Ls;                    // ||L.s||^2
            d  += Ls * cv;                    // (L.s) . c
            cc += cv * cv;                    // ||c||^2
        }
        // divisor in [8, 20] -> hw rcp is safe; avoids IEEE div sequence
        const float denom =
            0.5f * n1 - (d * d) * __builtin_amdgcn_rcpf(8.0f + 2.0f * cc);
        val = __builtin_amdgcn_rcpf(denom);   // v_rcp_f32
    }

    __shared__ float smem[32];
    const float bsum = block_reduce(val, smem);
    if (threadIdx.x == 0) partial[blockIdx.x] = bsum;
}

// Stage 2: deterministic final sum of block partials (single block).
__global__ void final_reduce_kernel(const float* __restrict__ partial, int n,
                                    float* __restrict__ out) {
    float v = 0.0f;
    for (int i = threadIdx.x; i < n; i += blockDim.x) v += partial[i];
    __shared__ float smem[32];
    const float total = block_reduce(v, smem);
    if (threadIdx.x == 0) out[0] = total;
}

extern "C" void kernel_launch(void* const* d_in, const int* in_sizes, int n_in,
                              void* d_out, int out_size, void* d_ws, size_t ws_size,
                              hipStream_t stream) {
    const float* x = (const float*)d_in[0];  // (B, 1024, 26) f32
    // d_in[1] (cond) and d_in[2] (time) are unused by the reference math,
    // but time's element count gives us B robustly.
    const int B = in_sizes[2];
    const int P = B * 26;

    const int threads = 256;                 // 8 waves per block (wave32)
    const int blocks  = (P + threads - 1) / threads;  // 416 for B=4096

    float* partial = (float*)d_ws;           // blocks * 4 bytes of scratch

    reflected_mass_kernel<<<blocks, threads, 0, stream>>>(x, partial, B);
    final_reduce_kernel<<<1, 256, 0, stream>>>(partial, blocks, (float*)d_out);
}